// ODESolver_18880676233843
// MI455X (gfx1250) — compile-verified
//
#include <hip/hip_runtime.h>
#include <hip/hip_bf16.h>

// ---------------------------------------------------------------------------
// Tsit5 neural-ODE solver for MI455X (gfx1250), full-f32 WMMA path.
//
// Roofline: 29 GFLOP total, ~20 MB total unique traffic -> both trivial for
// MI455X (23.3 TB/s, huge WMMA rates). The binding constraint is the
// 511*6 = 3066-deep serial chain of M=16 stage-GEMMs, so:
//   - exact-precision V_WMMA_F32_16X16X4_F32 (f16 rate buys nothing here,
//     f32 avoids 512-step error accumulation),
//   - W1^T resident in WGP LDS (pair-interleaved + XOR bank swizzle,
//     conflict-free single ds_load_b64 per fragment),
//   - W2^T fragments resident in VGPRs (128 VGPRs/lane, loaded once),
//   - all RK state (y, k1..k6) in VGPRs; only the 16x128 stage result
//     round-trips through a small LDS buffer,
//   - 3 barriers per stage, nontemporal output stores, prefetch of the
//     next step's input row.
// 4 workgroups x 256 threads (8 wave32), 16 trajectories each.
// ---------------------------------------------------------------------------

typedef __attribute__((ext_vector_type(2))) float v2f;
typedef __attribute__((ext_vector_type(8))) float v8f;

#define K_STEPS  512
#define N_STATE  128
#define N_INPUT  32
#define HIDDEN   256
#define FANIN    161           // 1 + N_STATE + N_INPUT
#define KPAD     164           // FANIN padded to multiple of 4 (WMMA K=4)
#define ZS       164           // Z row stride (floats)
#define HS       260           // H row stride (floats) - bank-conflict offset
#define KS       132           // k-stage row stride (floats)
#define MTILE    16            // trajectories per workgroup
#define NTHREADS 256           // 8 wave32

// LDS layout (floats).  B1 pair-interleaved: element (kk, n) lives at
//   (kk>>2)*1024 + ((kk>>1)&1)*512 + ((2*n + (kk&1)) ^ (((kk>>1)&1)*32))
#define B1_FLOATS  (KPAD / 4 * 1024)               // 41984
#define OFF_Z      B1_FLOATS                       // 41984
#define OFF_KST    (OFF_Z + MTILE * ZS)            // 44608
#define OFF_H      (OFF_KST + MTILE * KS)          // 46720
#define LDS_FLOATS (OFF_H + MTILE * HS)            // 50880
#define LDS_BYTES  (LDS_FLOATS * 4)                // 203520 (<320KB WGP LDS)

#if __has_builtin(__builtin_amdgcn_tanhf)
#define TANHF(x) __builtin_amdgcn_tanhf(x)
#else
#define TANHF(x) tanhf(x)
#endif

__launch_bounds__(NTHREADS, 1)
__global__ void tsit5_wmma_kernel(const float* __restrict__ ts,
                                  const float* __restrict__ y0,
                                  const float* __restrict__ us,
                                  const float* __restrict__ W1,
                                  const float* __restrict__ b1,
                                  const float* __restrict__ W2,
                                  const float* __restrict__ b2,
                                  float* __restrict__ out) {
  extern __shared__ float smem[];
  float* B1  = smem;            // pair-interleaved, swizzled W1^T (padded)
  float* Z   = smem + OFF_Z;    // [16][ZS]  stage input (t | ytmp | u(theta))
  float* KST = smem + OFF_KST;  // [16][KS]  stage output k
  float* H   = smem + OFF_H;    // [16][HS]  hidden activations

  const int tid    = threadIdx.x;
  const int lane   = tid & 31;
  const int wv     = tid >> 5;          // wave id 0..7
  const int ln     = lane & 15;         // N column within tile / M row for A
  const int gg     = lane >> 4;         // lane-group 0|1
  const int g2     = gg << 1;           // K sub-offset for A/B fragments
  const int g512   = gg << 9;           // B1 group sub-block
  const int gx     = gg << 5;           // B1 bank swizzle XOR
  const int rowoff = gg << 3;           // M offset for C/D fragments (0|8)
  const int b0     = blockIdx.x * MTILE;

  // ---- one-time: stage W1^T into LDS (pair-interleaved + swizzled) ----
  for (int idx = tid; idx < HIDDEN * FANIN; idx += NTHREADS) {
    int n = idx / FANIN;                // hidden unit (N)
    int f = idx - n * FANIN;            // fan-in index (K)
    int g = (f >> 1) & 1;
    B1[((f >> 2) << 10) + (g << 9) + ((2 * n + (f & 1)) ^ (g << 5))] = W1[idx];
  }
  for (int idx = tid; idx < (KPAD - FANIN) * HIDDEN; idx += NTHREADS) {
    int f = FANIN + idx / HIDDEN;       // zero K-padding rows 161..163
    int n = idx - (idx / HIDDEN) * HIDDEN;
    int g = (f >> 1) & 1;
    B1[((f >> 2) << 10) + (g << 9) + ((2 * n + (f & 1)) ^ (g << 5))] = 0.0f;
  }
  if (tid < MTILE * (ZS - FANIN)) {     // zero Z padding columns once
    int r = tid / (ZS - FANIN);
    int c = FANIN + tid % (ZS - FANIN);
    Z[r * ZS + c] = 0.0f;
  }

  // per-wave output-tile assignment and bias registers (never change)
  const int n1a = 16 * wv;              // layer-1 N-tiles: wv and wv+8
  const int n1b = 16 * (wv + 8);
  const int n2  = 16 * wv;              // layer-2 N-tile
  const float bias1a = b1[n1a + ln];
  const float bias1b = b1[n1b + ln];
  const float bias2  = b2[n2 + ln];

  // ---- W2^T fragments live in VGPRs for the whole solve ----
  // fragment kb, lane (gg,ln): { W2[n2+ln][4kb+2g], W2[n2+ln][4kb+2g+1] }
  v2f w2f[HIDDEN / 4];
#pragma unroll
  for (int kb = 0; kb < HIDDEN / 4; ++kb) {
    const int kk = (kb << 2) + g2;
    w2f[kb] = *(const v2f*)(W2 + (n2 + ln) * HIDDEN + kk);
  }

  // per-thread RK state: 8 consecutive flat (traj,state) elements in VGPRs
  const int ebase = tid << 3;
  const int traj  = ebase >> 7;         // 0..15
  const int st0   = ebase & 127;        // multiple of 8
  float y[8];
#pragma unroll
  for (int q = 0; q < 8; ++q) y[q] = y0[(b0 + traj) * N_STATE + st0 + q];
#pragma unroll
  for (int q = 0; q < 8; ++q)
    __builtin_nontemporal_store(
        y[q], out + ((size_t)(b0 + traj) * K_STEPS) * N_STATE + st0 + q);

  // per-thread u ownership: 2 consecutive inputs of one trajectory
  const int utraj = tid >> 4;
  const int uj    = (tid << 1) & 31;
  const float* usbase = us + ((size_t)(b0 + utraj) * K_STEPS) * N_INPUT + uj;

  // Tsit5 tableau (b7 = 0)
  const float CT[6] = {0.0f, 0.161f, 0.327f, 0.9f, 0.9800255409045097f, 1.0f};
  const float AT[6][5] = {
      {0.f, 0.f, 0.f, 0.f, 0.f},
      {0.161f, 0.f, 0.f, 0.f, 0.f},
      {-0.008480655492356989f, 0.335480655492357f, 0.f, 0.f, 0.f},
      {2.8971530571054935f, -6.359448489975075f, 4.3622954328695815f, 0.f, 0.f},
      {5.325864828439257f, -11.748883564062828f, 7.4955393428898365f,
       -0.09249506636175525f, 0.f},
      {5.86145544294642f, -12.92096931784711f, 8.159367898576159f,
       -0.071584973281401f, -0.028269050394068383f}};
  const float BT[6] = {0.09646076681806523f, 0.01f, 0.4798896504144996f,
                       1.379008574103742f, -3.290069515436081f,
                       2.324710524099774f};

  __syncthreads();

#pragma unroll 1
  for (int i = 0; i < K_STEPS - 1; ++i) {
    const float t0 = ts[i];
    const float hh = ts[i + 1] - t0;
    // backward-Hermite input interpolation data (loaded once per step)
    const float* urow = usbase + (size_t)i * N_INPUT;
    const float u0x = urow[0],        u0y = urow[1];
    const float u1x = urow[N_INPUT],  u1y = urow[N_INPUT + 1];
    __builtin_prefetch(urow + 2 * N_INPUT, 0, 1);     // next step's u row
    const float hd1x = u1x - u0x, hd1y = u1y - u0y;   // h * d[i+1]
    float hd0x, hd0y;                                 // h * d[i]
    if (i == 0) { hd0x = hd1x; hd0y = hd1y; }
    else {
      const float r = hh / (t0 - ts[i - 1]);
      hd0x = (u0x - urow[-N_INPUT]) * r;
      hd0y = (u0y - urow[1 - N_INPUT]) * r;
    }

    float k[6][8];
#pragma unroll
    for (int s = 0; s < 6; ++s) {
      // ---- build Z = [t | y + h*sum(a*k) | u(theta)] ----
      const float th = CT[s];
#pragma unroll
      for (int q = 0; q < 8; ++q) {
        float acc = y[q];
#pragma unroll
        for (int j = 0; j < 5; ++j)
          if (j < s) acc += (hh * AT[s][j]) * k[j][q];
        Z[traj * ZS + 1 + st0 + q] = acc;
      }
      if (tid < MTILE) Z[tid * ZS] = t0 + th * hh;
      {
        const float t2 = th * th, t3 = t2 * th;
        const float c00 = 2.f * t3 - 3.f * t2 + 1.f;
        const float c01 = t3 - 2.f * t2 + th;
        const float c10 = 3.f * t2 - 2.f * t3;
        const float c11 = t3 - t2;
        Z[utraj * ZS + 1 + N_STATE + uj] =
            c00 * u0x + c01 * hd0x + c10 * u1x + c11 * hd1x;
        Z[utraj * ZS + 1 + N_STATE + uj + 1] =
            c00 * u0y + c01 * hd0y + c10 * u1y + c11 * hd1y;
      }
      __syncthreads();                               // (1) Z ready

      // ---- layer 1: (16 x 164) @ (164 x 256), exact f32 WMMA ----
      v8f c1a, c1b;
#pragma unroll
      for (int r = 0; r < 8; ++r) { c1a[r] = bias1a; c1b[r] = bias1b; }
      for (int kb = 0; kb < KPAD / 4; ++kb) {
        const int kk = (kb << 2) + g2;
        v2f a = *(const v2f*)(Z + ln * ZS + kk);     // A[M=ln][K=kk,kk+1]
        v2f ba = *(const v2f*)(B1 + (kb << 10) + g512 +
                               (((n1a + ln) << 1) ^ gx));
        v2f bb = *(const v2f*)(B1 + (kb << 10) + g512 +
                               (((n1b + ln) << 1) ^ gx));
        c1a = __builtin_amdgcn_wmma_f32_16x16x4_f32(false, a, false, ba,
                                                    (short)0, c1a, false, false);
        c1b = __builtin_amdgcn_wmma_f32_16x16x4_f32(false, a, false, bb,
                                                    (short)0, c1b, false, false);
      }
#pragma unroll
      for (int r = 0; r < 8; ++r) {                  // tanh + stage to LDS
        H[(r + rowoff) * HS + n1a + ln] = TANHF(c1a[r]);
        H[(r + rowoff) * HS + n1b + ln] = TANHF(c1b[r]);
      }
      __syncthreads();                               // (2) H ready

      // ---- layer 2: (16 x 256) @ (256 x 128), W2 fragments in VGPRs ----
      v8f c2;
#pragma unroll
      for (int r = 0; r < 8; ++r) c2[r] = bias2;
#pragma unroll
      for (int kb = 0; kb < HIDDEN / 4; ++kb) {
        const int kk = (kb << 2) + g2;
        v2f a = *(const v2f*)(H + ln * HS + kk);
        c2 = __builtin_amdgcn_wmma_f32_16x16x4_f32(false, a, false, w2f[kb],
                                                   (short)0, c2, false, false);
      }
#pragma unroll
      for (int r = 0; r < 8; ++r)
        KST[(r + rowoff) * KS + n2 + ln] = c2[r];
      __syncthreads();                               // (3) k ready

      // pull this stage's k into per-thread registers; no trailing barrier:
      // next stage's barrier (1) orders these reads vs. the next KST writes.
#pragma unroll
      for (int q = 0; q < 8; ++q) k[s][q] = KST[traj * KS + st0 + q];
    }

    // ---- y_{n+1} = y + h * sum(b_j * k_j), store trajectory sample ----
#pragma unroll
    for (int q = 0; q < 8; ++q) {
      float acc = y[q];
#pragma unroll
      for (int s = 0; s < 6; ++s) acc += (hh * BT[s]) * k[s][q];
      y[q] = acc;
      __builtin_nontemporal_store(
          acc, out + ((size_t)(b0 + traj) * K_STEPS + (i + 1)) * N_STATE +
                   st0 + q);
    }
  }
}

extern "C" void kernel_launch(void* const* d_in, const int* in_sizes, int n_in,
                              void* d_out, int out_size, void* d_ws,
                              size_t ws_size, hipStream_t stream) {
  (void)in_sizes; (void)n_in; (void)out_size; (void)d_ws; (void)ws_size;
  const float* ts = (const float*)d_in[0];
  const float* y0 = (const float*)d_in[1];
  const float* us = (const float*)d_in[2];
  const float* W1 = (const float*)d_in[3];
  const float* b1 = (const float*)d_in[4];
  const float* W2 = (const float*)d_in[5];
  const float* b2 = (const float*)d_in[6];
  float* out = (float*)d_out;

  hipFuncSetAttribute((const void*)tsit5_wmma_kernel,
                      hipFuncAttributeMaxDynamicSharedMemorySize, LDS_BYTES);
  tsit5_wmma_kernel<<<dim3(64 / MTILE), dim3(NTHREADS), LDS_BYTES, stream>>>(
      ts, y0, us, W1, b1, W2, b2, out);
}